// LLELoss_5634997093006
// MI455X (gfx1250) — compile-verified
//
#include <hip/hip_runtime.h>
#include <hip/hip_bf16.h>
#include <math.h>

typedef __attribute__((ext_vector_type(2))) float v2f;
typedef __attribute__((ext_vector_type(8))) float v8f;

#define N_PTS 8192
#define DIM   512
#define ZDIM  64
#define KNN   10
#define REGL  1e-6f
#define TILE  16
#define MTILES 2                 // A tiles per block -> 32 rows
#define ROWS  (TILE * MTILES)    // 32
#define NWAVES 8
#define NTILES (N_PTS / TILE)    // 512
#define NBLOCKS (N_PTS / ROWS)   // 256
#define A_STRIDE 516             // 512 + 4 pad -> conflict-free ds_load_b64 frags

// ---------------- kernel 0: row squared norms ----------------
__global__ __launch_bounds__(256) void sqnorm_kernel(const float* __restrict__ X,
                                                     float* __restrict__ sq) {
  int wave = threadIdx.x >> 5;
  int lane = threadIdx.x & 31;
  int row = blockIdx.x * NWAVES + wave;
  const float* xr = X + (size_t)row * DIM;
  float s = 0.f;
  for (int d = lane; d < DIM; d += 32) { float v = xr[d]; s += v * v; }
  #pragma unroll
  for (int off = 16; off > 0; off >>= 1) s += __shfl_xor(s, off, 32);
  if (lane == 0) sq[row] = s;
}

// ---------------- kernel 1: fused Gram + streaming kNN ----------------
// One block per 32-row stripe (two 16-row A tiles in LDS). 8 waves split the
// 512 column tiles in pairs; per k-step each wave issues 2 LDS A-frags +
// 2 global B-frags feeding 4 WMMAs (0.5 global + 0.5 LDS load per WMMA).
__global__ __launch_bounds__(256) void knn_kernel(const float* __restrict__ X,
                                                  const float* __restrict__ sq,
                                                  int* __restrict__ nbr) {
  __shared__ float As[ROWS * A_STRIDE];           // 64.5 KB
  __shared__ float tileBuf[NWAVES][TILE * 17];    //  8.5 KB
  __shared__ float candD[ROWS][NWAVES * 20];      // 20.0 KB
  __shared__ int   candI[ROWS][NWAVES * 20];      // 20.0 KB

  const int I    = blockIdx.x;
  const int tid  = threadIdx.x;
  const int wave = tid >> 5;
  const int lane = tid & 31;
  const int m    = lane & 15;   // row (A) / col (B) within tile
  const int g    = lane >> 4;   // half-wave K/column-group select

  // stage both A tiles into LDS (padded stride)
  for (int idx = tid; idx < ROWS * DIM; idx += 256) {
    int r = idx >> 9, c = idx & 511;
    As[r * A_STRIDE + c] = X[((size_t)(I * ROWS + r)) * DIM + c];
  }
  __syncthreads();

  // two per-lane top-10 lists (one per A tile), sorted desc (bd[0] = worst)
  float bd0[KNN], bd1[KNN]; int bi0[KNN], bi1[KNN];
  #pragma unroll
  for (int j = 0; j < KNN; ++j) {
    bd0[j] = 3.0e38f; bi0[j] = 0;
    bd1[j] = 3.0e38f; bi1[j] = 0;
  }

  const float* Abase0 = As + m * A_STRIDE + (g << 1);
  const float* Abase1 = Abase0 + TILE * A_STRIDE;
  const int gRow0 = I * ROWS + m;
  const int gRow1 = gRow0 + TILE;

  // dump one 16x16 score tile and fold its candidates into a top-10 list
  auto phase = [&](const v8f& acc, float (&bd)[KNN], int (&bi)[KNN],
                   int Jcur, int gRow) {
    float sqc = sq[Jcur * TILE + m];
    #pragma unroll
    for (int r = 0; r < 8; ++r)
      tileBuf[wave][(r + 8 * g) * 17 + m] = sqc - 2.0f * acc[r];
    __builtin_amdgcn_wave_barrier();
    #pragma unroll
    for (int c0 = 0; c0 < 8; ++c0) {
      int c = (g << 3) + c0;
      int col = Jcur * TILE + c;
      float s = tileBuf[wave][m * 17 + c];
      if (col != gRow && s < bd[0]) {
        bd[0] = s; bi[0] = col;
        #pragma unroll
        for (int j = 0; j < KNN - 1; ++j) {
          if (bd[j] < bd[j + 1]) {
            float td = bd[j]; bd[j] = bd[j + 1]; bd[j + 1] = td;
            int   ti = bi[j]; bi[j] = bi[j + 1]; bi[j + 1] = ti;
          }
        }
      }
    }
    __builtin_amdgcn_wave_barrier();
  };

  for (int Jt = wave * 2; Jt < NTILES; Jt += NWAVES * 2) {
    v8f acc00 = {}, acc01 = {}, acc10 = {}, acc11 = {};
    const float* B0 = X + ((size_t)(Jt * TILE + m)) * DIM + (g << 1);
    const float* B1 = B0 + (size_t)TILE * DIM;
    #pragma unroll 4
    for (int k = 0; k < DIM; k += 4) {
      v2f a0 = *(const v2f*)(Abase0 + k);   // ds_load_b64 (conflict-free)
      v2f a1 = *(const v2f*)(Abase1 + k);
      v2f b0 = *(const v2f*)(B0 + k);       // global (L2-resident)
      v2f b1 = *(const v2f*)(B1 + k);
      acc00 = __builtin_amdgcn_wmma_f32_16x16x4_f32(false, a0, false, b0,
                                                    (short)0, acc00, false, false);
      acc01 = __builtin_amdgcn_wmma_f32_16x16x4_f32(false, a0, false, b1,
                                                    (short)0, acc01, false, false);
      acc10 = __builtin_amdgcn_wmma_f32_16x16x4_f32(false, a1, false, b0,
                                                    (short)0, acc10, false, false);
      acc11 = __builtin_amdgcn_wmma_f32_16x16x4_f32(false, a1, false, b1,
                                                    (short)0, acc11, false, false);
    }
    phase(acc00, bd0, bi0, Jt,     gRow0);
    phase(acc01, bd0, bi0, Jt + 1, gRow0);
    phase(acc10, bd1, bi1, Jt,     gRow1);
    phase(acc11, bd1, bi1, Jt + 1, gRow1);
  }

  // dump per-lane lists, merge across waves
  #pragma unroll
  for (int j = 0; j < KNN; ++j) {
    candD[m][wave * 20 + g * 10 + j]        = bd0[j];
    candI[m][wave * 20 + g * 10 + j]        = bi0[j];
    candD[TILE + m][wave * 20 + g * 10 + j] = bd1[j];
    candI[TILE + m][wave * 20 + g * 10 + j] = bi1[j];
  }
  __syncthreads();
  if (tid < ROWS) {
    float fd[KNN]; int fi[KNN];
    #pragma unroll
    for (int j = 0; j < KNN; ++j) { fd[j] = 3.0e38f; fi[j] = 0; }
    for (int e = 0; e < NWAVES * 20; ++e) {
      float s = candD[tid][e];
      int col = candI[tid][e];
      if (s < fd[0]) {
        fd[0] = s; fi[0] = col;
        #pragma unroll
        for (int j = 0; j < KNN - 1; ++j) {
          if (fd[j] < fd[j + 1]) {
            float td = fd[j]; fd[j] = fd[j + 1]; fd[j + 1] = td;
            int   ti = fi[j]; fi[j] = fi[j + 1]; fi[j + 1] = ti;
          }
        }
      }
    }
    #pragma unroll
    for (int j = 0; j < KNN; ++j)
      nbr[((size_t)(I * ROWS + tid)) * KNN + j] = fi[j];
  }
}

// ---------------- kernel 2: per-point local solve + recon error ----------------
__global__ __launch_bounds__(32) void lle_point_kernel(const float* __restrict__ X,
                                                       const float* __restrict__ Z,
                                                       const int* __restrict__ nbr,
                                                       float* __restrict__ err) {
  __shared__ float sdiff[KNN][DIM];   // 20 KB
  __shared__ float sC[KNN * KNN];
  __shared__ float sw[KNN];
  __shared__ int   snb[KNN];
  const int i = blockIdx.x;
  const int lane = threadIdx.x;

  if (lane < KNN) snb[lane] = nbr[(size_t)i * KNN + lane];
  __syncthreads();

  const float* xi = X + (size_t)i * DIM;
  for (int k = 0; k < KNN; ++k) {
    const float* xn = X + (size_t)snb[k] * DIM;
    for (int d = lane; d < DIM; d += 32)
      sdiff[k][d] = xn[d] - xi[d];
  }
  __syncthreads();

  // C = diff @ diff^T (symmetric, 55 dots, wave-reduced)
  for (int a = 0; a < KNN; ++a)
    for (int b = a; b < KNN; ++b) {
      float s = 0.f;
      for (int d = lane; d < DIM; d += 32) s += sdiff[a][d] * sdiff[b][d];
      #pragma unroll
      for (int off = 16; off > 0; off >>= 1) s += __shfl_xor(s, off, 32);
      if (lane == 0) {
        if (a == b) s += REGL;
        sC[a * KNN + b] = s;
        sC[b * KNN + a] = s;
      }
    }
  __syncthreads();

  if (lane == 0) {
    float rhs[KNN], w[KNN];
    #pragma unroll
    for (int j = 0; j < KNN; ++j) rhs[j] = 1.0f;
    #pragma unroll
    for (int p = 0; p < KNN; ++p) {
      float inv = 1.0f / sC[p * KNN + p];
      #pragma unroll
      for (int r2 = 0; r2 < KNN; ++r2) {
        if (r2 > p) {
          float f = sC[r2 * KNN + p] * inv;
          #pragma unroll
          for (int c = 0; c < KNN; ++c)
            if (c >= p) sC[r2 * KNN + c] -= f * sC[p * KNN + c];
          rhs[r2] -= f * rhs[p];
        }
      }
    }
    #pragma unroll
    for (int p = KNN - 1; p >= 0; --p) {
      float s = rhs[p];
      #pragma unroll
      for (int c = 0; c < KNN; ++c)
        if (c > p) s -= sC[p * KNN + c] * w[c];
      w[p] = s / sC[p * KNN + p];
    }
    float sum = 0.f;
    #pragma unroll
    for (int j = 0; j < KNN; ++j) sum += w[j];
    float invs = 1.0f / sum;
    #pragma unroll
    for (int j = 0; j < KNN; ++j) sw[j] = w[j] * invs;
  }
  __syncthreads();

  // reconstruction error over 64 dims (2 per lane)
  float e = 0.f;
  #pragma unroll
  for (int dd = 0; dd < 2; ++dd) {
    int d = lane + dd * 32;
    float r = 0.f;
    #pragma unroll
    for (int k = 0; k < KNN; ++k) r += sw[k] * Z[(size_t)snb[k] * ZDIM + d];
    float t = r - Z[(size_t)i * ZDIM + d];
    e += t * t;
  }
  #pragma unroll
  for (int off = 16; off > 0; off >>= 1) e += __shfl_xor(e, off, 32);
  if (lane == 0) err[i] = e;
}

// ---------------- kernel 3: deterministic mean reduction ----------------
__global__ __launch_bounds__(256) void reduce_kernel(const float* __restrict__ err,
                                                     float* __restrict__ out) {
  __shared__ float buf[256];
  float s = 0.f;
  for (int i = threadIdx.x; i < N_PTS; i += 256) s += err[i];
  buf[threadIdx.x] = s;
  __syncthreads();
  for (int off = 128; off > 0; off >>= 1) {
    if (threadIdx.x < off) buf[threadIdx.x] += buf[threadIdx.x + off];
    __syncthreads();
  }
  if (threadIdx.x == 0) out[0] = buf[0] / (float)((size_t)N_PTS * ZDIM);
}

extern "C" void kernel_launch(void* const* d_in, const int* in_sizes, int n_in,
                              void* d_out, int out_size, void* d_ws, size_t ws_size,
                              hipStream_t stream) {
  const float* X = (const float*)d_in[0];   // 8192 x 512 f32
  const float* Z = (const float*)d_in[1];   // 8192 x 64  f32
  float* out = (float*)d_out;               // scalar f32

  char* ws = (char*)d_ws;
  float* sq  = (float*)ws;                                   // 32 KB
  int*   nbr = (int*)(ws + 32 * 1024);                       // 320 KB
  float* err = (float*)(ws + 32 * 1024 + N_PTS * KNN * 4);   // 32 KB

  sqnorm_kernel<<<N_PTS / NWAVES, 256, 0, stream>>>(X, sq);
  knn_kernel<<<NBLOCKS, 256, 0, stream>>>(X, sq, nbr);
  lle_point_kernel<<<N_PTS, 32, 0, stream>>>(X, Z, nbr, err);
  reduce_kernel<<<1, 256, 0, stream>>>(err, out);
}